// GraphSAGEClassifier_53163105190232
// MI455X (gfx1250) — compile-verified
//
#include <hip/hip_runtime.h>
#include <hip/hip_bf16.h>

// GraphSAGE forward for MI455X (gfx1250, wave32).
// Strategy: everything (h buffers ~17MB) is L2-resident; GEMMs run on the
// f32 WMMA pipe (V_WMMA_F32_16X16X4_F32) to preserve reference numerics.
// Conv layer computes h@W_top + agg@W_bot (split concat) in one accumulator.

typedef __attribute__((ext_vector_type(2))) float v2f;
typedef __attribute__((ext_vector_type(8))) float v8f;

#define GN 4096
#define GH 256
#define GB 4
#define GM (GB * GN)   // 16384 rows

// ---------------------------------------------------------------------------
// WMMA GEMM: out[M x 256] = act(A1[M x K1] @ W1[K1 x 256]
//                               (+ A2[M x K2] @ W2[K2 x 256]) + bias)
// Block: 256 threads = 8 waves as 4(M) x 2(N) -> 64 x 32 output tile.
// K staged in 32-wide chunks through LDS; W stored transposed in LDS so both
// A and B wave fragments are contiguous 8-byte (v2f) reads.
// ---------------------------------------------------------------------------
__global__ __launch_bounds__(256)
void wmma_gemm(const float* __restrict__ A1, const float* __restrict__ W1,
               const float* __restrict__ A2, const float* __restrict__ W2,
               const float* __restrict__ bias, float* __restrict__ out,
               int K1, int K2, int do_relu) {
    __shared__ float As[64 * 32];    // 8 KB: A tile, row-major [64][32]
    __shared__ float BsT[32 * 32];   // 4 KB: W tile transposed [n=32][k=32]

    const int tid     = threadIdx.x;
    const int lane    = tid & 31;
    const int waveId  = tid >> 5;          // 0..7
    const int wm      = waveId >> 1;       // 0..3 (M slice)
    const int wn      = waveId & 1;        // 0..1 (N slice)
    const int rowBase = blockIdx.x * 64;
    const int colBase = blockIdx.y * 32;

    // fragment lane decomposition (ISA 7.12.2, 32-bit 16x4 A / mirrored B)
    const int mrow  = lane & 15;
    const int khalf = lane >> 4;           // 0 or 1 -> k offset 0 or 2
    const int kh2   = khalf * 2;
    const int aRowOff = (wm * 16 + mrow) * 32;
    const int bColOff = (wn * 16 + mrow) * 32;   // n = lane&15, same decomposition

    v8f acc = {};   // 16x16 f32 accumulator (8 VGPRs)

    for (int op = 0; op < 2; ++op) {
        const float* A = (op == 0) ? A1 : A2;
        const float* W = (op == 0) ? W1 : W2;
        const int    K = (op == 0) ? K1 : K2;
        if (K <= 0) continue;              // uniform branch, EXEC stays full

        for (int k0 = 0; k0 < K; k0 += 32) {
            __syncthreads();
            {   // A tile: 64x32 = 512 float4, 2 per thread
                const int row = tid >> 3;            // 0..31
                const int c4  = (tid & 7) << 2;      // 0,4,...,28
                *(float4*)&As[row * 32 + c4] =
                    *(const float4*)&A[(size_t)(rowBase + row) * K + k0 + c4];
                *(float4*)&As[(row + 32) * 32 + c4] =
                    *(const float4*)&A[(size_t)(rowBase + row + 32) * K + k0 + c4];
            }
            {   // W tile: 32(k)x32(n), store transposed [n][k]
                const int kr = tid >> 3;             // 0..31
                const int c4 = (tid & 7) << 2;       // 0,4,...,28
                float4 w = *(const float4*)&W[(size_t)(k0 + kr) * GH + colBase + c4];
                BsT[(c4 + 0) * 32 + kr] = w.x;
                BsT[(c4 + 1) * 32 + kr] = w.y;
                BsT[(c4 + 2) * 32 + kr] = w.z;
                BsT[(c4 + 3) * 32 + kr] = w.w;
            }
            __syncthreads();

            #pragma unroll
            for (int kk = 0; kk < 32; kk += 4) {
                v2f a = *(const v2f*)&As[aRowOff + kk + kh2];
                v2f b = *(const v2f*)&BsT[bColOff + kk + kh2];
                acc = __builtin_amdgcn_wmma_f32_16x16x4_f32(
                        /*neg_a=*/false, a, /*neg_b=*/false, b,
                        /*c_mod=*/(short)0, acc,
                        /*reuse_a=*/false, /*reuse_b=*/false);
            }
        }
    }

    // Epilogue: D VGPR r -> (M = r + 8*khalf, N = lane&15) per ISA layout.
    const int col = colBase + wn * 16 + (lane & 15);
    const float bv = bias[col];
    #pragma unroll
    for (int r = 0; r < 8; ++r) {
        const int row = rowBase + wm * 16 + r + 8 * khalf;
        float v = acc[r] + bv;
        if (do_relu) v = fmaxf(v, 0.0f);
        out[(size_t)row * GH + col] = v;
    }
}

// ---------------------------------------------------------------------------
// Neighbor aggregation: agg[b,n,:] = mean over valid neighbors of h[b,idx,:].
// One block per node, 256 threads = one per channel; gathers are coalesced
// 1KB rows served out of L2 (h is L2-resident).
// ---------------------------------------------------------------------------
__global__ __launch_bounds__(256)
void aggregate_kernel(const float* __restrict__ h,
                      const long long* __restrict__ neigh,
                      float* __restrict__ agg) {
    __shared__ int   s_idx[32];
    __shared__ float s_inv;
    const int n   = blockIdx.x;
    const int tid = threadIdx.x;

    if (tid < 32) {
        long long id = neigh[(size_t)n * 32 + tid];
        s_idx[tid] = (id >= 0) ? (int)id : -1;
    }
    __syncthreads();
    if (tid == 0) {
        int cnt = 0;
        #pragma unroll
        for (int d = 0; d < 32; ++d) cnt += (s_idx[d] >= 0) ? 1 : 0;
        s_inv = 1.0f / (float)(cnt > 0 ? cnt : 1);
    }
    __syncthreads();

    const float inv = s_inv;
    #pragma unroll
    for (int b = 0; b < GB; ++b) {
        float sum = 0.0f;
        for (int d = 0; d < 32; ++d) {
            const int id = s_idx[d];                 // uniform per iteration
            if (id >= 0)
                sum += h[((size_t)b * GN + id) * GH + tid];
        }
        agg[((size_t)b * GN + n) * GH + tid] = sum * inv;
    }
}

// ---------------------------------------------------------------------------
// Readout: g[b,:] = sum_n h[b,n,:]  (chunked, f32 atomics into zeroed g)
// ---------------------------------------------------------------------------
__global__ void zero_g_kernel(float* __restrict__ g) {
    g[blockIdx.x * 256 + threadIdx.x] = 0.0f;
}

__global__ __launch_bounds__(256)
void readout_sum_kernel(const float* __restrict__ h, float* __restrict__ g) {
    const int b     = blockIdx.y;
    const int base  = blockIdx.x * 256;   // 16 chunks of 256 nodes
    const int tid   = threadIdx.x;
    float sum = 0.0f;
    for (int i = 0; i < 256; ++i)
        sum += h[((size_t)b * GN + base + i) * GH + tid];
    atomicAdd(&g[b * GH + tid], sum);
}

// ---------------------------------------------------------------------------
// MLP head: out = relu(g @ W_r1 + b_r1) @ W_r2 + b_r2   (4x10 result)
// ---------------------------------------------------------------------------
__global__ __launch_bounds__(256)
void head_kernel(const float* __restrict__ g,
                 const float* __restrict__ W1, const float* __restrict__ b1,
                 const float* __restrict__ W2, const float* __restrict__ b2,
                 float* __restrict__ out) {
    __shared__ float s_g[GB * GH];
    __shared__ float s_r[GB * GH];
    const int tid = threadIdx.x;

    #pragma unroll
    for (int b = 0; b < GB; ++b) s_g[b * GH + tid] = g[b * GH + tid];
    __syncthreads();

    #pragma unroll
    for (int b = 0; b < GB; ++b) {
        float acc = b1[tid];
        for (int k = 0; k < GH; ++k)
            acc += s_g[b * GH + k] * W1[k * GH + tid];
        s_r[b * GH + tid] = fmaxf(acc, 0.0f);
    }
    __syncthreads();

    if (tid < GB * 10) {
        const int b = tid / 10, c = tid % 10;
        float acc = b2[c];
        for (int k = 0; k < GH; ++k)
            acc += s_r[b * GH + k] * W2[k * 10 + c];
        out[b * 10 + c] = acc;
    }
}

// ---------------------------------------------------------------------------
extern "C" void kernel_launch(void* const* d_in, const int* in_sizes, int n_in,
                              void* d_out, int out_size, void* d_ws, size_t ws_size,
                              hipStream_t stream) {
    const float*     x       = (const float*)d_in[0];
    const long long* neigh   = (const long long*)d_in[1];
    const float*     W_in    = (const float*)d_in[2];
    const float*     b_in    = (const float*)d_in[3];
    const float*     W_convs = (const float*)d_in[4];
    const float*     b_convs = (const float*)d_in[5];
    const float*     W_r1    = (const float*)d_in[6];
    const float*     b_r1    = (const float*)d_in[7];
    const float*     W_r2    = (const float*)d_in[8];
    const float*     b_r2    = (const float*)d_in[9];
    float* out = (float*)d_out;

    float* ws = (float*)d_ws;
    const size_t HSZ = (size_t)GM * GH;   // 4,194,304 floats (16 MB)
    float* hA  = ws;
    float* hB  = ws + HSZ;
    float* agg = ws + 2 * HSZ;
    float* g   = ws + 3 * HSZ;            // 1024 floats

    const dim3 gemmGrid(GM / 64, GH / 32);   // 256 x 8 blocks

    // input projection: h = x @ W_in + b_in  (no activation)
    wmma_gemm<<<gemmGrid, 256, 0, stream>>>(x, W_in, nullptr, nullptr,
                                            b_in, hA, 128, 0, 0);

    float* cur = hA;
    float* nxt = hB;
    for (int l = 0; l < 3; ++l) {
        aggregate_kernel<<<GN, 256, 0, stream>>>(cur, neigh, agg);
        const float* Wt = W_convs + (size_t)l * (2 * GH) * GH;  // top half (h)
        const float* Wb = Wt + (size_t)GH * GH;                 // bottom half (agg)
        wmma_gemm<<<gemmGrid, 256, 0, stream>>>(cur, Wt, agg, Wb,
                                                b_convs + l * GH, nxt,
                                                GH, GH, 1);
        float* t = cur; cur = nxt; nxt = t;
    }

    zero_g_kernel<<<GB, 256, 0, stream>>>(g);
    readout_sum_kernel<<<dim3(GN / 256, GB), 256, 0, stream>>>(cur, g);
    head_kernel<<<1, 256, 0, stream>>>(g, W_r1, b_r1, W_r2, b_r2, out);
}